// ClassificationPointTransformer_46909632807291
// MI455X (gfx1250) — compile-verified
//
#include <hip/hip_runtime.h>
#include <hip/hip_bf16.h>
#include <math.h>

// ============================================================================
// PointTransformer classifier for MI455X (gfx1250, wave32).
// fp32 end-to-end; dense GEMMs use V_WMMA_F32_16X16X4_F32 (exact fp32 matrix
// pipe). Attention conv is fused per-point in LDS (avoids materializing the
// [B,n,17,c] delta tensor). Whole working set (~45MB ws) lives in L2 (192MB).
// ============================================================================

typedef float v2f __attribute__((ext_vector_type(2)));
typedef float v8f __attribute__((ext_vector_type(8)));

#define BATCH 16

// ---------------------------------------------------------------------------
// GEMM: C[M,N] = A[M,K] @ W[K,N] (+bias) (+relu). One wave per 16x16 tile.
// f32 WMMA fragment layouts (CDNA5 ISA):
//   A 16x4 : lanes 0-15 -> M=lane, v0=K0 v1=K1 ; lanes 16-31 -> M=lane-16, v0=K2 v1=K3
//   B 4x16 : mirrored (v0: K0 / K2 per half, v1: K1 / K3), N = lane&15
//   D 16x16: vgpr r -> M = r + 8*(lane>>4), N = lane&15
// Main loop is unguarded (K%4==0 for all call sites but mlp_in); the K=3 case
// is handled by a single guarded tail so the hot loop has no divergent exec
// masking. A-fragment loads are an aligned float2 (row*K even when Kmain>0).
// ---------------------------------------------------------------------------
template <bool RELU>
__global__ __launch_bounds__(128) void gemm_kernel(
    const float* __restrict__ A, const float* __restrict__ W,
    const float* __restrict__ bias, float* __restrict__ C,
    int M, int N, int K, int totalTiles)
{
    int wave = blockIdx.x * (blockDim.x >> 5) + (threadIdx.x >> 5);
    if (wave >= totalTiles) return;               // wave-uniform: EXEC all-1 inside
    int tilesN = N >> 4;
    int tm = wave / tilesN, tn = wave % tilesN;
    int lane = threadIdx.x & 31;
    int half = lane >> 4;
    int l15  = lane & 15;
    int row  = tm * 16 + l15;
    int col  = tn * 16 + l15;

    const float* Arow = A + (size_t)row * K;
    v8f acc = {0.f, 0.f, 0.f, 0.f, 0.f, 0.f, 0.f, 0.f};

    int Kmain = K & ~3;
    for (int k0 = 0; k0 < Kmain; k0 += 4) {
        int ka = k0 + (half << 1);
        v2f a = *(const v2f*)(Arow + ka);          // 8B-aligned: K even, ka even
        v2f b;
        b.x = W[(size_t)(ka    ) * N + col];
        b.y = W[(size_t)(ka + 1) * N + col];
#if defined(__gfx1250__)
        acc = __builtin_amdgcn_wmma_f32_16x16x4_f32(
            false, a, false, b, (short)0, acc, false, false);
#else
        (void)a; (void)b;
#endif
    }
    if (Kmain < K) {                               // scalar-uniform tail (K=3 only)
        int ka = Kmain + (half << 1);
        v2f a, b;
        a.x = (ka     < K) ? Arow[ka    ] : 0.f;
        a.y = (ka + 1 < K) ? Arow[ka + 1] : 0.f;
        b.x = (ka     < K) ? W[(size_t)(ka    ) * N + col] : 0.f;
        b.y = (ka + 1 < K) ? W[(size_t)(ka + 1) * N + col] : 0.f;
#if defined(__gfx1250__)
        acc = __builtin_amdgcn_wmma_f32_16x16x4_f32(
            false, a, false, b, (short)0, acc, false, false);
#else
        (void)a; (void)b;
#endif
    }

    float bv = bias ? bias[col] : 0.f;
    int mbase = tm * 16 + (half << 3);
#pragma unroll
    for (int r = 0; r < 8; ++r) {
        float v = acc[r] + bv;
        if (RELU) v = fmaxf(v, 0.f);
        C[(size_t)(mbase + r) * N + col] = v;
    }
}

// ---------------------------------------------------------------------------
// BatchNorm batch statistics: one block per channel. Biased variance (jnp.var).
// stats[c] = mean, stats[C+c] = 1/sqrt(var+eps)
// ---------------------------------------------------------------------------
__global__ __launch_bounds__(256) void bn_stats_kernel(
    const float* __restrict__ X, int M, int C, float* __restrict__ stats)
{
    int c = blockIdx.x;
    __shared__ float ss[256], sq[256];
    float s = 0.f, q = 0.f;
    for (int m = threadIdx.x; m < M; m += 256) {
        float v = X[m * C + c];
        s += v; q += v * v;
    }
    ss[threadIdx.x] = s; sq[threadIdx.x] = q;
    __syncthreads();
    for (int st = 128; st > 0; st >>= 1) {
        if (threadIdx.x < st) {
            ss[threadIdx.x] += ss[threadIdx.x + st];
            sq[threadIdx.x] += sq[threadIdx.x + st];
        }
        __syncthreads();
    }
    if (threadIdx.x == 0) {
        float mu  = ss[0] / (float)M;
        float var = sq[0] / (float)M - mu * mu;
        stats[c]     = mu;
        stats[C + c] = rsqrtf(var + 1e-5f);
    }
}

__global__ void bn_apply_relu_kernel(
    const float* __restrict__ X, const float* __restrict__ stats,
    const float* __restrict__ g, const float* __restrict__ be,
    float* __restrict__ Y, int M, int C)
{
    int t = blockIdx.x * blockDim.x + threadIdx.x;
    if (t >= M * C) return;
    int c = t % C;
    float v = (X[t] - stats[c]) * stats[C + c] * g[c] + be[c];
    Y[t] = fmaxf(v, 0.f);
}

// ---------------------------------------------------------------------------
// Brute-force kNN, KK nearest (incl. self when qpos==cpos). Register-resident
// sorted list via unrolled bubble-insert; ties keep lowest index (== top_k).
// ---------------------------------------------------------------------------
template <int KK>
__global__ __launch_bounds__(128) void knn_kernel(
    const float* __restrict__ qpos, const float* __restrict__ cpos,
    int nq, int nc, int* __restrict__ nbr)
{
    int t = blockIdx.x * blockDim.x + threadIdx.x;
    if (t >= BATCH * nq) return;
    int b = t / nq, i = t % nq;
    const float* q  = qpos + (size_t)(b * nq + i) * 3;
    const float* cb = cpos + (size_t)b * nc * 3;
    float qx = q[0], qy = q[1], qz = q[2];
    float bd[KK]; int bi[KK];
#pragma unroll
    for (int s = 0; s < KK; ++s) { bd[s] = 3.4e38f; bi[s] = 0; }
    for (int j = 0; j < nc; ++j) {
        float dx = qx - cb[j * 3 + 0];
        float dy = qy - cb[j * 3 + 1];
        float dz = qz - cb[j * 3 + 2];
        float d = dx * dx + dy * dy + dz * dz;
        if (d < bd[KK - 1]) {
            float cd = d; int ci = j;
#pragma unroll
            for (int s = 0; s < KK; ++s) {       // strict <: equal dists keep earlier j
                if (cd < bd[s]) {
                    float td = bd[s]; int ti = bi[s];
                    bd[s] = cd; bi[s] = ci;
                    cd = td; ci = ti;
                }
            }
        }
    }
    int* o = nbr + (size_t)(b * nq + i) * KK;
#pragma unroll
    for (int s = 0; s < KK; ++s) o[s] = bi[s];
}

// ---------------------------------------------------------------------------
// Farthest point sampling: one block (256 thr) per batch element. Deterministic
// start at 0; argmax tie-break = first occurrence (matches jnp.argmax).
// ---------------------------------------------------------------------------
__global__ __launch_bounds__(256) void fps_kernel(
    const float* __restrict__ pos, int n, int n_sub, int* __restrict__ idx)
{
    int b = blockIdx.x;
    __shared__ float dist[2048];
    __shared__ float rv[256];
    __shared__ int   ri[256];
    __shared__ int   s_last;
    const float* pb = pos + (size_t)b * n * 3;
    for (int t = threadIdx.x; t < n; t += 256) dist[t] = 3.4e38f;
    if (threadIdx.x == 0) { idx[b * n_sub] = 0; s_last = 0; }
    __syncthreads();
    for (int step = 1; step < n_sub; ++step) {
        int last = s_last;
        float lx = pb[last * 3 + 0], ly = pb[last * 3 + 1], lz = pb[last * 3 + 2];
        float bv = -1.f; int bi = 0x7fffffff;
        for (int t = threadIdx.x; t < n; t += 256) {
            float dx = pb[t * 3 + 0] - lx;
            float dy = pb[t * 3 + 1] - ly;
            float dz = pb[t * 3 + 2] - lz;
            float d  = dx * dx + dy * dy + dz * dz;
            float nd = fminf(dist[t], d);
            dist[t] = nd;
            if (nd > bv) { bv = nd; bi = t; }    // strict >: first max wins per thread
        }
        rv[threadIdx.x] = bv; ri[threadIdx.x] = bi;
        __syncthreads();
        for (int st = 128; st > 0; st >>= 1) {
            if (threadIdx.x < st) {
                bool take = (rv[threadIdx.x + st] > rv[threadIdx.x]) ||
                            (rv[threadIdx.x + st] == rv[threadIdx.x] &&
                             ri[threadIdx.x + st] <  ri[threadIdx.x]);
                if (take) { rv[threadIdx.x] = rv[threadIdx.x + st];
                            ri[threadIdx.x] = ri[threadIdx.x + st]; }
            }
            __syncthreads();
        }
        if (threadIdx.x == 0) { s_last = ri[0]; idx[b * n_sub + step] = ri[0]; }
        __syncthreads();
    }
}

__global__ void gather_pos_kernel(const float* __restrict__ pos,
                                  const int* __restrict__ idx,
                                  float* __restrict__ sub, int n, int ns)
{
    int t = blockIdx.x * blockDim.x + threadIdx.x;
    if (t >= BATCH * ns * 3) return;
    int d = t % 3, r = t / 3, b = r / ns, i = r % ns;
    sub[t] = pos[(size_t)(b * n + idx[b * ns + i]) * 3 + d];
}

__global__ void gather_max_kernel(const float* __restrict__ h,
                                  const int* __restrict__ nbr,
                                  float* __restrict__ out,
                                  int n, int ns, int C, int kk)
{
    int t = blockIdx.x * blockDim.x + threadIdx.x;
    if (t >= BATCH * ns * C) return;
    int ch = t % C, r = t / C, b = r / ns, i = r % ns;
    const int* nb = nbr + (size_t)(b * ns + i) * kk;
    float m = -3.4e38f;
    for (int j = 0; j < kk; ++j)
        m = fmaxf(m, h[(size_t)(b * n + nb[j]) * C + ch]);
    out[t] = m;
}

__global__ void meanpool_kernel(const float* __restrict__ h,
                                float* __restrict__ g, int n, int C)
{
    int t = blockIdx.x * blockDim.x + threadIdx.x;
    if (t >= BATCH * C) return;
    int b = t / C, ch = t % C;
    float s = 0.f;
    for (int i = 0; i < n; ++i) s += h[(size_t)(b * n + i) * C + ch];
    g[t] = s / (float)n;
}

// ---------------------------------------------------------------------------
// Fused PointTransformerConv: one block (256 thr) per point.
//   delta_ij = MLP2(pos_i - pos_j)            (3 -> 64 -> C, relu/relu)
//   alpha_ij = MLP2(adst_i - asrc_j + delta)  (C -> 64 -> C, relu/relu)
//   w = softmax_j(alpha) ; out_i = sum_j w * (v_j + delta_ij)
// Phase A stores alpha[kk][C] in LDS; softmax normalized in place; phase B
// recomputes delta (cheaper than 17*C more LDS) and accumulates.
// LDS: 17*C + C + 128 floats  (C=512 -> ~37 KB)
// ---------------------------------------------------------------------------
template <int C>
__global__ __launch_bounds__(256) void tpconv_kernel(
    const float* __restrict__ pos, const int* __restrict__ nbr,
    const float* __restrict__ asrc, const float* __restrict__ adst,
    const float* __restrict__ vmat,
    const float* __restrict__ pW1, const float* __restrict__ pb1,
    const float* __restrict__ pW2, const float* __restrict__ pb2,
    const float* __restrict__ aW1, const float* __restrict__ ab1,
    const float* __restrict__ aW2, const float* __restrict__ ab2,
    float* __restrict__ out, int n, int kk)
{
    __shared__ float s_alpha[17 * C];
    __shared__ float s_adst[C];
    __shared__ float s_t1[64];
    __shared__ float s_t2[64];

    int p = blockIdx.x;
    int b = p / n, i = p % n;
    int tid = threadIdx.x;
    const float* pb = pos + (size_t)b * n * 3;
    const int*   nb = nbr + (size_t)p * kk;
    float pix = pb[i * 3 + 0], piy = pb[i * 3 + 1], piz = pb[i * 3 + 2];

    for (int ch = tid; ch < C; ch += 256) s_adst[ch] = adst[(size_t)p * C + ch];
    __syncthreads();

    // ---- phase A: alpha for every neighbor ----
    for (int jj = 0; jj < kk; ++jj) {
        int j = nb[jj];
        float dx = pix - pb[j * 3 + 0];
        float dy = piy - pb[j * 3 + 1];
        float dz = piz - pb[j * 3 + 2];
        if (tid < 64) {
            float t = pb1[tid] + dx * pW1[tid] + dy * pW1[64 + tid] + dz * pW1[128 + tid];
            s_t1[tid] = fmaxf(t, 0.f);
        }
        __syncthreads();
        for (int ch = tid; ch < C; ch += 256) {
            float d = pb2[ch];
            for (int u = 0; u < 64; ++u) d += s_t1[u] * pW2[u * C + ch];
            d = fmaxf(d, 0.f);
            s_alpha[jj * C + ch] = s_adst[ch] - asrc[(size_t)(b * n + j) * C + ch] + d;
        }
        __syncthreads();
        if (tid < 64) {
            float t = ab1[tid];
            for (int ch = 0; ch < C; ++ch) t += s_alpha[jj * C + ch] * aW1[ch * 64 + tid];
            s_t2[tid] = fmaxf(t, 0.f);
        }
        __syncthreads();
        for (int ch = tid; ch < C; ch += 256) {
            float a2 = ab2[ch];
            for (int u = 0; u < 64; ++u) a2 += s_t2[u] * aW2[u * C + ch];
            s_alpha[jj * C + ch] = fmaxf(a2, 0.f);
        }
        __syncthreads();
    }

    // ---- stable softmax over neighbors, normalized weights in place ----
    for (int ch = tid; ch < C; ch += 256) {
        float m = -3.4e38f;
        for (int jj = 0; jj < kk; ++jj) m = fmaxf(m, s_alpha[jj * C + ch]);
        float s = 0.f;
        for (int jj = 0; jj < kk; ++jj) s += expf(s_alpha[jj * C + ch] - m);
        float inv = 1.f / s;
        for (int jj = 0; jj < kk; ++jj)
            s_alpha[jj * C + ch] = expf(s_alpha[jj * C + ch] - m) * inv;
    }

    // ---- phase B: weighted sum with recomputed delta ----
    constexpr int NQ = (C + 255) / 256;
    float acc[NQ];
#pragma unroll
    for (int q = 0; q < NQ; ++q) acc[q] = 0.f;
    for (int jj = 0; jj < kk; ++jj) {
        int j = nb[jj];
        float dx = pix - pb[j * 3 + 0];
        float dy = piy - pb[j * 3 + 1];
        float dz = piz - pb[j * 3 + 2];
        __syncthreads();                          // protect s_t1 reuse
        if (tid < 64) {
            float t = pb1[tid] + dx * pW1[tid] + dy * pW1[64 + tid] + dz * pW1[128 + tid];
            s_t1[tid] = fmaxf(t, 0.f);
        }
        __syncthreads();
        int q = 0;
        for (int ch = tid; ch < C; ch += 256, ++q) {
            float d = pb2[ch];
            for (int u = 0; u < 64; ++u) d += s_t1[u] * pW2[u * C + ch];
            d = fmaxf(d, 0.f);
            acc[q] += s_alpha[jj * C + ch] * (vmat[(size_t)(b * n + j) * C + ch] + d);
        }
    }
    {
        int q = 0;
        for (int ch = tid; ch < C; ch += 256, ++q)
            out[(size_t)p * C + ch] = acc[q];
    }
}

// ---------------------------------------------------------------------------
// Head: g[16,512] -> relu(@W1[512,64]+b1) -> @W2[64,40]+b2 -> softmax. 1 block.
// ---------------------------------------------------------------------------
__global__ __launch_bounds__(1024) void head_kernel(
    const float* __restrict__ g,
    const float* __restrict__ W1, const float* __restrict__ b1,
    const float* __restrict__ W2, const float* __restrict__ b2,
    float* __restrict__ out)
{
    __shared__ float t1[16 * 64];
    __shared__ float lg[16 * 40];
    int tid = threadIdx.x;
    if (tid < 16 * 64) {
        int r = tid >> 6, u = tid & 63;
        float s = b1[u];
        for (int c = 0; c < 512; ++c) s += g[r * 512 + c] * W1[c * 64 + u];
        t1[tid] = fmaxf(s, 0.f);
    }
    __syncthreads();
    if (tid < 16 * 40) {
        int r = tid / 40, o = tid % 40;
        float s = b2[o];
        for (int u = 0; u < 64; ++u) s += t1[r * 64 + u] * W2[u * 40 + o];
        lg[tid] = s;
    }
    __syncthreads();
    if (tid < 16) {
        float m = -3.4e38f;
        for (int o = 0; o < 40; ++o) m = fmaxf(m, lg[tid * 40 + o]);
        float s = 0.f;
        for (int o = 0; o < 40; ++o) s += expf(lg[tid * 40 + o] - m);
        float inv = 1.f / s;
        for (int o = 0; o < 40; ++o) out[tid * 40 + o] = expf(lg[tid * 40 + o] - m) * inv;
    }
}

// ===========================================================================
// Host orchestration.
// d_in layout assumption: [0]=x, [1]=pos, then params pytree flattened with
// sorted dict keys (jax convention):
//   blocks[0..4] x 15: Wdst,Win,Wout,Wsrc,Wv,aW1,aW2,ab1,ab2,bin,bout,pW1,pW2,pb1,pb2
//   mlp_in: W,b,be,g   mlp_out: W1,W2,b1,b2   td[0..3] x 4: W,b,be,g
// ===========================================================================
extern "C" void kernel_launch(void* const* d_in, const int* in_sizes, int n_in,
                              void* d_out, int out_size, void* d_ws, size_t ws_size,
                              hipStream_t stream)
{
    (void)in_sizes; (void)n_in; (void)out_size; (void)ws_size;
    const int dims[5] = {32, 64, 128, 256, 512};

    const float* x      = (const float*)d_in[0];
    const float* pos_in = (const float*)d_in[1];
    auto P = [&](int i) -> const float* { return (const float*)d_in[i]; };
    const int PB = 2;                 // blocks start
    const int MLPIN  = PB + 5 * 15;   // 77: W,b,be,g
    const int MLPOUT = MLPIN + 4;     // 81: W1,W2,b1,b2
    const int TD     = MLPOUT + 4;    // 85: per-i W,b,be,g
    // block leaf offsets (sorted keys)
    const int BWdst = 0, BWin = 1, BWout = 2, BWsrc = 3, BWv = 4;
    const int BaW1 = 5, BaW2 = 6, Bab1 = 7, Bab2 = 8, Bbin = 9, Bbout = 10;
    const int BpW1 = 11, BpW2 = 12, Bpb1 = 13, Bpb2 = 14;

    // ---- workspace bump allocator ----
    char* base = (char*)d_ws;
    size_t off = 0;
    auto alloc = [&](size_t nbytes) -> void* {
        void* p = base + off;
        off += (nbytes + 255) & ~(size_t)255;
        return p;
    };
    const size_t MAXF = (size_t)BATCH * 2048 * 32;       // largest per-stage feature map
    float* fH   = (float*)alloc(MAXF * 4);
    float* fHin = (float*)alloc(MAXF * 4);
    float* fV   = (float*)alloc(MAXF * 4);
    float* fAs  = (float*)alloc(MAXF * 4);
    float* fAd  = (float*)alloc(MAXF * 4);
    float* fCv  = (float*)alloc(MAXF * 4);
    float* fTD  = (float*)alloc((size_t)BATCH * 2048 * 64 * 4);  // td linear @ full res
    float* posA = (float*)alloc((size_t)BATCH * 2048 * 3 * 4);
    float* posB = (float*)alloc((size_t)BATCH * 2048 * 3 * 4);
    int*   nbr  = (int*)alloc((size_t)BATCH * 2048 * 17 * 4);
    int*   nbrT = (int*)alloc((size_t)BATCH * 512 * 16 * 4);
    int*   idxS = (int*)alloc((size_t)BATCH * 512 * 4);
    float* stats= (float*)alloc(2 * 512 * 4);
    float* gpool= (float*)alloc((size_t)BATCH * 512 * 4);

    auto gemm = [&](const float* A, const float* W, const float* bias, float* Cc,
                    int M, int N, int K, bool relu) {
        int tiles = (M / 16) * (N / 16);
        int blocks = (tiles + 3) / 4;
        if (relu) gemm_kernel<true ><<<blocks, 128, 0, stream>>>(A, W, bias, Cc, M, N, K, tiles);
        else      gemm_kernel<false><<<blocks, 128, 0, stream>>>(A, W, bias, Cc, M, N, K, tiles);
    };
    auto bn = [&](float* X, int M, int C, const float* g, const float* be, float* Y) {
        bn_stats_kernel<<<C, 256, 0, stream>>>(X, M, C, stats);
        int tot = M * C;
        bn_apply_relu_kernel<<<(tot + 255) / 256, 256, 0, stream>>>(X, stats, g, be, Y, M, C);
    };
    auto conv = [&](int c, int s, const float* pos, const int* nb, int n, int kk, float* outp) {
        int bs = PB + 15 * s;
        const float *pW1 = P(bs + BpW1), *pb1 = P(bs + Bpb1);
        const float *pW2 = P(bs + BpW2), *pb2 = P(bs + Bpb2);
        const float *aW1 = P(bs + BaW1), *ab1 = P(bs + Bab1);
        const float *aW2 = P(bs + BaW2), *ab2 = P(bs + Bab2);
        dim3 grid(BATCH * n);
        switch (c) {
        case 32:  tpconv_kernel<32 ><<<grid, 256, 0, stream>>>(pos, nb, fAs, fAd, fV, pW1, pb1, pW2, pb2, aW1, ab1, aW2, ab2, outp, n, kk); break;
        case 64:  tpconv_kernel<64 ><<<grid, 256, 0, stream>>>(pos, nb, fAs, fAd, fV, pW1, pb1, pW2, pb2, aW1, ab1, aW2, ab2, outp, n, kk); break;
        case 128: tpconv_kernel<128><<<grid, 256, 0, stream>>>(pos, nb, fAs, fAd, fV, pW1, pb1, pW2, pb2, aW1, ab1, aW2, ab2, outp, n, kk); break;
        case 256: tpconv_kernel<256><<<grid, 256, 0, stream>>>(pos, nb, fAs, fAd, fV, pW1, pb1, pW2, pb2, aW1, ab1, aW2, ab2, outp, n, kk); break;
        case 512: tpconv_kernel<512><<<grid, 256, 0, stream>>>(pos, nb, fAs, fAd, fV, pW1, pb1, pW2, pb2, aW1, ab1, aW2, ab2, outp, n, kk); break;
        }
    };
    auto knn_self = [&](const float* pos, int n, int kk) {
        int thr = BATCH * n, blocks = (thr + 127) / 128;
        if (kk == 17)     knn_kernel<17><<<blocks, 128, 0, stream>>>(pos, pos, n, n, nbr);
        else /* kk==8 */  knn_kernel< 8><<<blocks, 128, 0, stream>>>(pos, pos, n, n, nbr);
    };
    auto tblock = [&](int s, int c, int n, const float* pos, const int* nb, int kk) {
        int bs = PB + 15 * s;
        int M = BATCH * n;
        gemm(fH,   P(bs + BWin),  P(bs + Bbin),  fHin, M, c, c, true);   // lin_in + relu
        gemm(fHin, P(bs + BWv),   nullptr,       fV,   M, c, c, false);  // value
        gemm(fHin, P(bs + BWsrc), nullptr,       fAs,  M, c, c, false);  // alpha_j
        gemm(fHin, P(bs + BWdst), nullptr,       fAd,  M, c, c, false);  // alpha_i
        conv(c, s, pos, nb, n, kk, fCv);
        gemm(fCv,  P(bs + BWout), P(bs + Bbout), fH,   M, c, c, true);   // lin_out + relu
    };

    // ---- mlp_in: relu(bn(x @ W + b)) ----
    gemm(x, P(MLPIN + 0), P(MLPIN + 1), fTD, BATCH * 2048, 32, 3, false);
    bn(fTD, BATCH * 2048, 32, P(MLPIN + 3), P(MLPIN + 2), fH);

    // ---- stage 0 ----
    int n = 2048, c = 32;
    const float* posCur = pos_in;
    knn_self(posCur, n, 17);
    tblock(0, c, n, posCur, nbr, 17);

    // ---- 4x (TransitionDown + TransformerBlock) ----
    float* posArena[2] = {posA, posB};
    for (int i = 0; i < 4; ++i) {
        int ns = n / 4, cn = dims[i + 1];
        fps_kernel<<<BATCH, 256, 0, stream>>>(posCur, n, ns, idxS);
        float* posNext = posArena[i & 1];
        {
            int thr = BATCH * ns * 3;
            gather_pos_kernel<<<(thr + 255) / 256, 256, 0, stream>>>(posCur, idxS, posNext, n, ns);
        }
        {
            int thr = BATCH * ns, blocks = (thr + 127) / 128;
            knn_kernel<16><<<blocks, 128, 0, stream>>>(posNext, posCur, ns, n, nbrT);
        }
        int tb = TD + 4 * i;   // W,b,be,g
        gemm(fH, P(tb + 0), P(tb + 1), fTD, BATCH * n, cn, c, false);
        bn(fTD, BATCH * n, cn, P(tb + 3), P(tb + 2), fTD);     // in-place elementwise
        {
            int thr = BATCH * ns * cn;
            gather_max_kernel<<<(thr + 255) / 256, 256, 0, stream>>>(fTD, nbrT, fH, n, ns, cn, 16);
        }
        posCur = posNext; n = ns; c = cn;
        int kk = (n < 17) ? n : 17;                            // n=8 at last stage
        knn_self(posCur, n, kk);
        tblock(i + 1, c, n, posCur, nbr, kk);
    }

    // ---- global mean pool + head ----
    {
        int thr = BATCH * 512;
        meanpool_kernel<<<(thr + 255) / 256, 256, 0, stream>>>(fH, gpool, n, 512);
    }
    head_kernel<<<1, 1024, 0, stream>>>(gpool, P(MLPOUT + 0), P(MLPOUT + 2),
                                        P(MLPOUT + 1), P(MLPOUT + 3), (float*)d_out);
}